// EDUA_1872605741607
// MI455X (gfx1250) — compile-verified
//
#include <hip/hip_runtime.h>
#include <math.h>

typedef __attribute__((ext_vector_type(2))) float v2f;
typedef __attribute__((ext_vector_type(8))) float v8f;
typedef __attribute__((ext_vector_type(2))) int   v2i;

#define D     64
#define LLEN  200
#define SLOTS 31
#define BATCH 1024

// ---------------------------------------------------------------------------
// Projection GEMM with f32 WMMA:  out[r, :] = table[gidx[r], :] @ W  (W is [64,64],
// contracted over W's FIRST index, matching einsum 'be,ed->bd').
// One wave (32 threads) computes one 16x16 output tile, K-loop of 16 steps of
// V_WMMA_F32_16X16X4_F32.
//
// ISA lane layouts (CDNA5 ISA):
//   A 16x4 f32 : lanes 0-15 -> M=lane, {v0,v1}={K0,K1}; lanes 16-31 -> {K2,K3}
//   C 16x16 f32: vgpr r -> (M = r + (lane>>4)*8, N = lane&15)
//   B 4x16 f32 : (row-striped across lanes) v0 = {K0 | K2}, v1 = {K1 | K3},
//                N = lane&15  (mirrors A's K split across lane halves)
// ---------------------------------------------------------------------------
__global__ void proj_wmma_kernel(const float* __restrict__ table,
                                 const int*   __restrict__ gidx,
                                 const float* __restrict__ W,
                                 float*       __restrict__ out,
                                 int nrows) {
  const int lane = threadIdx.x;          // 0..31 (wave32)
  const int tileM = blockIdx.x;          // 16-row tile
  const int n0   = blockIdx.y * 16;      // 16-col tile
  const int mlo  = lane & 15;
  const int khi  = (lane >> 4) * 2;      // 0 or 2

  int row = tileM * 16 + mlo;
  int src = (row < nrows) ? gidx[row] : 0;
  const float* arow = table + (long)src * D;

  v8f c = {};
  #pragma unroll
  for (int k = 0; k < D; k += 4) {
    v2f a, b;
    a.x = arow[k + khi + 0];
    a.y = arow[k + khi + 1];
    b.x = W[(k + khi + 0) * D + n0 + mlo];
    b.y = W[(k + khi + 1) * D + n0 + mlo];
    // 8 args: (neg_a, A, neg_b, B, c_mod, C, reuse_a, reuse_b)
    c = __builtin_amdgcn_wmma_f32_16x16x4_f32(false, a, false, b,
                                              (short)0, c, false, false);
  }
  const int mbase = tileM * 16 + (lane >> 4) * 8;
  #pragma unroll
  for (int r = 0; r < 8; ++r) {
    int m = mbase + r;
    if (m < nrows) out[(long)m * D + n0 + mlo] = c[r];
  }
}

// ---------------------------------------------------------------------------
// Masked-softmax attention over one neighbor list, nq queries sharing the list.
// One block per batch element. Gather the [200,64] list into LDS ONCE, reuse it
// for score pass + weighted-sum pass.
//
// CDNA5 data path: the row gather uses GLOBAL_LOAD_ASYNC_TO_LDS_B64 (per-lane
// global addresses -> per-lane LDS addresses, tracked by ASYNCcnt) so the DMA
// writes LDS directly without a VGPR round-trip; s_wait_asynccnt 0 before the
// block barrier. Falls back to load+ds_store if the builtin is unavailable.
// ---------------------------------------------------------------------------
#if defined(__has_builtin)
#if __has_builtin(__builtin_amdgcn_global_load_async_to_lds_b64)
#define USE_ASYNC_LDS 1
#endif
#endif

#ifdef USE_ASYNC_LDS
typedef __attribute__((address_space(1))) v2i* gptr_v2i;
typedef __attribute__((address_space(3))) v2i* lptr_v2i;
#endif

__device__ __forceinline__ void wait_asynccnt0() {
#if defined(__has_builtin) && __has_builtin(__builtin_amdgcn_s_wait_asynccnt)
  __builtin_amdgcn_s_wait_asynccnt(0);
#else
  asm volatile("s_wait_asynccnt 0" ::: "memory");
#endif
}

__global__ void attend_kernel(const int*   __restrict__ idx,    // [B, L] (pre-offset for per-n lists)
                              const float* __restrict__ table,  // [NE, 64]
                              const float* __restrict__ qw,     // projected queries
                              int qw_bstride,                   // floats per b in qw
                              int nq,                           // queries per b (stride 64)
                              float*       __restrict__ out,    // [B, SLOTS, 64]
                              int slot0) {
  extern __shared__ float smem[];
  float* E   = smem;              // [L*64] gathered rows
  float* sc  = E + LLEN * D;      // [256] scores / weights
  float* qb  = sc + 256;          // [<=256] queries
  float* red = qb + 256;          // [256] reduction scratch
  __shared__ int ids[LLEN];

  const int b    = blockIdx.x;
  const int tid  = threadIdx.x;   // 256 threads = 8 waves
  const int lane = tid & 31;
  const int wid  = tid >> 5;

  const int* lb = idx + (long)b * LLEN;
#ifdef USE_ASYNC_LDS
  // Async DMA gather: each lane moves 8B of a row; 32 lanes cover one 256B row.
  for (int l = wid; l < LLEN; l += 8) {
    int id = lb[l];
    if (lane == 0) ids[l] = id;
    const float* src = table + (long)id * D + lane * 2;
    float* dst = E + l * D + lane * 2;
    __builtin_amdgcn_global_load_async_to_lds_b64(
        (gptr_v2i)((v2i*)src),
        (lptr_v2i)((v2i*)dst),
        /*offset=*/0, /*cpol=*/0);
  }
#else
  for (int l = wid; l < LLEN; l += 8) {
    int id = lb[l];
    if (lane == 0) ids[l] = id;
    const float2 v = *(const float2*)(table + (long)id * D + lane * 2);
    E[l * D + lane * 2 + 0] = v.x;
    E[l * D + lane * 2 + 1] = v.y;
  }
#endif

  // queries -> LDS (overlaps with outstanding async gather)
  for (int t = tid; t < nq * D; t += blockDim.x)
    qb[t] = qw[(long)b * qw_bstride + t];

#ifdef USE_ASYNC_LDS
  wait_asynccnt0();
#endif
  __syncthreads();

  for (int q = 0; q < nq; ++q) {
    const float* qv = qb + q * D;
    float s = -__builtin_inff();
    if (tid < LLEN && ids[tid] != 0) {
      const float* er = E + tid * D;
      const int off = tid & 63;          // rotate -> conflict-free LDS reads
      float acc = 0.f;
      #pragma unroll 8
      for (int d0 = 0; d0 < D; ++d0) {
        int d = (d0 + off) & 63;
        acc += er[d] * qv[d];
      }
      s = acc;
    }
    sc[tid] = s;
    __syncthreads();
    for (int st = 128; st >= 1; st >>= 1) {        // block max
      if (tid < st) { float o = sc[tid + st]; if (o > sc[tid]) sc[tid] = o; }
      __syncthreads();
    }
    const float mx = sc[0];
    __syncthreads();
    const float e = (tid < LLEN && s > -__builtin_inff()) ? __expf(s - mx) : 0.f;
    sc[tid] = e;                                    // unnormalized weights
    red[tid] = e;
    __syncthreads();
    for (int st = 128; st >= 1; st >>= 1) {        // block sum
      if (tid < st) red[tid] += red[tid + st];
      __syncthreads();
    }
    const float inv = 1.f / red[0];
    __syncthreads();

    // weighted sum: thread (d, g) accumulates rows l = g, g+4, ...
    const int d = tid & 63, g = tid >> 6;
    float acc = 0.f;
    for (int l = g; l < LLEN; l += 4) acc += sc[l] * E[l * D + d];
    red[tid] = acc;
    __syncthreads();
    if (tid < D) {
      float o = red[tid] + red[tid + 64] + red[tid + 128] + red[tid + 192];
      out[((long)b * SLOTS + slot0 + q) * D + d] = o * inv;
    }
    __syncthreads();
  }
}

// ---------------------------------------------------------------------------
// Aspect reparameterization: mu + 0.01*exp(sg/2)*N(0,1), deterministic PCG noise.
// ---------------------------------------------------------------------------
__device__ __forceinline__ unsigned pcg(unsigned x) {
  x = x * 747796405u + 2891336453u;
  x = ((x >> ((x >> 28) + 4u)) ^ x) * 277803737u;
  return (x >> 22) ^ x;
}

__global__ void aspect_kernel(const float* __restrict__ asp,   // [nrows, 8]
                              const float* __restrict__ miu,   // [8, 64]
                              const float* __restrict__ sigma, // [8, 64]
                              float*       __restrict__ out,
                              int slot0, int nper, int nrows, unsigned seed) {
  const int t = blockIdx.x * blockDim.x + threadIdx.x;
  const int row = t >> 6, d = t & 63;
  if (row >= nrows) return;
  const float* a = asp + (long)row * 8;
  float mu = 0.f, sg = 0.f;
  #pragma unroll
  for (int m = 0; m < 8; ++m) {
    mu += a[m] * miu[m * D + d];
    sg += a[m] * sigma[m * D + d];
  }
  mu *= 0.125f; sg *= 0.125f;
  const unsigned h1 = pcg(seed ^ (unsigned)(t * 2654435761u));
  const unsigned h2 = pcg(h1 + 0x9e3779b9u);
  const float u1 = ((h1 >> 8) + 1.0f) * (1.0f / 16777217.0f);
  const float u2 = (h2 >> 8) * (1.0f / 16777216.0f);
  const float nrm = sqrtf(-2.f * __logf(u1)) * __cosf(6.2831853f * u2);
  const int bb = row / nper, r = row % nper;
  out[((long)bb * SLOTS + slot0 + r) * D + d] = mu + 0.01f * __expf(0.5f * sg) * nrm;
}

// ---------------------------------------------------------------------------
extern "C" void kernel_launch(void* const* d_in, const int* in_sizes, int n_in,
                              void* d_out, int out_size, void* d_ws, size_t ws_size,
                              hipStream_t stream) {
  const float* emb_user_mlp1 = (const float*)d_in[0];
  const float* emb_item_mlp1 = (const float*)d_in[1];
  const float* emb_user_mf1  = (const float*)d_in[2];
  const float* emb_item_mf1  = (const float*)d_in[3];
  const float* emb_user_mlp2 = (const float*)d_in[4];
  const float* emb_item_mlp2 = (const float*)d_in[5];
  const float* emb_user_mf2  = (const float*)d_in[6];
  const float* emb_item_mf2  = (const float*)d_in[7];
  const float* W_i1 = (const float*)d_in[8];
  const float* W_i2 = (const float*)d_in[9];
  const float* W_u1 = (const float*)d_in[10];
  const float* W_u2 = (const float*)d_in[11];
  const float* user_aspect_miu   = (const float*)d_in[12];
  const float* user_aspect_sigma = (const float*)d_in[13];
  const float* item_aspect_miu   = (const float*)d_in[14];
  const float* item_aspect_sigma = (const float*)d_in[15];
  const int* user_indices     = (const int*)d_in[16];
  const int* item_indices     = (const int*)d_in[17];
  const int* item_negs        = (const int*)d_in[18];  // [B,4]
  const int* rev_item_indices = (const int*)d_in[19];
  const int* rev_item_negs    = (const int*)d_in[20];
  const int* u_item_list      = (const int*)d_in[21];  // [B,200]
  const int* i_user_list      = (const int*)d_in[22];
  const int* neg_user_list    = (const int*)d_in[23];  // [4,B,200]
  const int* rev_i_user_list  = (const int*)d_in[24];
  const int* rev_neg_user_list= (const int*)d_in[25];
  const float* user_aspect        = (const float*)d_in[27];
  const float* item_aspect        = (const float*)d_in[28];
  const float* neg_item_aspect    = (const float*)d_in[29]; // [B,4,8]
  const float* reverse_item_aspect= (const float*)d_in[30];
  const float* reverse_neg_aspect = (const float*)d_in[31];
  float* out = (float*)d_out;

  // workspace: projected queries
  float* ws = (float*)d_ws;
  float* qW_i1  = ws;                    // 1024*64
  float* qW_u1  = qW_i1  + BATCH * D;
  float* qW_i2  = qW_u1  + BATCH * D;
  float* qW_u2  = qW_i2  + BATCH * D;
  float* qWn_i1 = qW_u2  + BATCH * D;    // 4096*64
  float* qWn_i2 = qWn_i1 + BATCH * 4 * D;

  // --- projections (WMMA) ---
  dim3 pb(32);
  dim3 g1(BATCH / 16, 4), g4(BATCH * 4 / 16, 4);
  proj_wmma_kernel<<<g1, pb, 0, stream>>>(emb_item_mlp1, item_indices,     W_i1, qW_i1,  BATCH);
  proj_wmma_kernel<<<g1, pb, 0, stream>>>(emb_user_mlp1, user_indices,     W_u1, qW_u1,  BATCH);
  proj_wmma_kernel<<<g1, pb, 0, stream>>>(emb_item_mlp2, rev_item_indices, W_i2, qW_i2,  BATCH);
  proj_wmma_kernel<<<g1, pb, 0, stream>>>(emb_user_mlp2, user_indices,     W_u2, qW_u2,  BATCH);
  proj_wmma_kernel<<<g4, pb, 0, stream>>>(emb_item_mlp1, item_negs,        W_i1, qWn_i1, BATCH * 4);
  proj_wmma_kernel<<<g4, pb, 0, stream>>>(emb_item_mlp2, rev_item_negs,    W_i2, qWn_i2, BATCH * 4);

  // --- attentions ---
  const size_t shmem = (size_t)(LLEN * D + 256 + 256 + 256) * sizeof(float);
  dim3 ab(256), ag(BATCH);
  // singles: slots 0..3
  attend_kernel<<<ag, ab, shmem, stream>>>(u_item_list,     emb_item_mf1, qW_i1, D, 1, out, 0);
  attend_kernel<<<ag, ab, shmem, stream>>>(i_user_list,     emb_user_mf1, qW_u1, D, 1, out, 1);
  attend_kernel<<<ag, ab, shmem, stream>>>(u_item_list,     emb_item_mf2, qW_i2, D, 1, out, 2);
  attend_kernel<<<ag, ab, shmem, stream>>>(rev_i_user_list, emb_user_mf2, qW_u2, D, 1, out, 3);
  // u_neg_rel: shared list, 4 queries -> slots 7..10
  attend_kernel<<<ag, ab, shmem, stream>>>(u_item_list, emb_item_mf1, qWn_i1, 4 * D, 4, out, 7);
  // neg_u_rel: per-n lists, shared query -> slots 11..14
  for (int n = 0; n < 4; ++n)
    attend_kernel<<<ag, ab, shmem, stream>>>(neg_user_list + (long)n * BATCH * LLEN,
                                             emb_user_mf1, qW_u1, D, 1, out, 11 + n);
  // ru_neg_rel -> slots 15..18
  attend_kernel<<<ag, ab, shmem, stream>>>(u_item_list, emb_item_mf2, qWn_i2, 4 * D, 4, out, 15);
  // rneg_u_rel -> slots 19..22
  for (int n = 0; n < 4; ++n)
    attend_kernel<<<ag, ab, shmem, stream>>>(rev_neg_user_list + (long)n * BATCH * LLEN,
                                             emb_user_mf2, qW_u2, D, 1, out, 19 + n);

  // --- aspects: slots 4,5,6 and 23..26, 27..30 ---
  dim3 sb(256);
  dim3 sg1((BATCH * D + 255) / 256), sg4((BATCH * 4 * D + 255) / 256);
  aspect_kernel<<<sg1, sb, 0, stream>>>(user_aspect,         user_aspect_miu, user_aspect_sigma, out, 4,  1, BATCH,     0x1111u);
  aspect_kernel<<<sg1, sb, 0, stream>>>(item_aspect,         item_aspect_miu, item_aspect_sigma, out, 5,  1, BATCH,     0x2222u);
  aspect_kernel<<<sg1, sb, 0, stream>>>(reverse_item_aspect, item_aspect_miu, item_aspect_sigma, out, 6,  1, BATCH,     0x3333u);
  aspect_kernel<<<sg4, sb, 0, stream>>>(neg_item_aspect,     item_aspect_miu, item_aspect_sigma, out, 23, 4, BATCH * 4, 0x4444u);
  aspect_kernel<<<sg4, sb, 0, stream>>>(reverse_neg_aspect,  item_aspect_miu, item_aspect_sigma, out, 27, 4, BATCH * 4, 0x5555u);

  (void)in_sizes; (void)n_in; (void)out_size; (void)ws_size;
}